// MemoryMapUpdater_34230889349758
// MI455X (gfx1250) — compile-verified
//
#include <hip/hip_runtime.h>
#include <hip/hip_bf16.h>

typedef float v2f __attribute__((ext_vector_type(2)));
typedef float v4f __attribute__((ext_vector_type(4)));
typedef float v8f __attribute__((ext_vector_type(8)));
typedef int   v4i __attribute__((ext_vector_type(4)));

#define NB_N 100000
#define NB_D 128
#define NB_C 1024
#define NB_NODES 1000000
#define ROWS_PB 32                  // rows of X per workgroup
#define COLS_PI 64                  // cols (map cells) per iteration
#define ITERS   (NB_C / COLS_PI)    // 16
#define LDS_STRIDE 132              // padded row stride (floats): conflict-free b64

#ifndef __has_builtin
#define __has_builtin(x) 0
#endif
#if __has_builtin(__builtin_amdgcn_global_load_async_to_lds_b128) && \
    __has_builtin(__builtin_amdgcn_s_wait_asynccnt)
#define USE_ASYNC_LDS 1
#else
#define USE_ASYNC_LDS 0
#endif

// ---------------------------------------------------------------------------
// Stage one 64x128 slab of memory_map into an LDS buffer (padded layout).
// Async (GLOBAL_LOAD_ASYNC_TO_LDS_B128, ASYNCcnt) when available, else
// synchronous v4 copy. Visibility is established by the caller's
// wait+barrier at the top of the next iteration (double buffering).
// ---------------------------------------------------------------------------
__device__ __forceinline__ void stage_m_tile(const float* __restrict__ mmap,
                                             float* __restrict__ dstbase,
                                             int c0, int tid) {
#if USE_ASYNC_LDS
#pragma unroll
  for (int i = 0; i < 8; ++i) {
    int idx = tid + i * 256;            // 0..2047 float4 slots
    int r = idx >> 5, c4 = idx & 31;
    const float* g = mmap + (size_t)(c0 + r) * NB_D + c4 * 4;
    float* l = dstbase + r * LDS_STRIDE + c4 * 4;
    __builtin_amdgcn_global_load_async_to_lds_b128((v4i*)g, (v4i*)l, 0, 0);
  }
#else
#pragma unroll
  for (int i = 0; i < 8; ++i) {
    int idx = tid + i * 256;
    int r = idx >> 5, c4 = idx & 31;
    v4f v = *(const v4f*)(mmap + (size_t)(c0 + r) * NB_D + c4 * 4);
    *(v4f*)(dstbase + r * LDS_STRIDE + c4 * 4) = v;
  }
#endif
}

__device__ __forceinline__ void wait_stage_done() {
#if USE_ASYNC_LDS
  __builtin_amdgcn_s_wait_asynccnt(0);
#endif
  __syncthreads();
}

// ---------------------------------------------------------------------------
// Init: copy node tables into output (full overwrite => deterministic under
// graph replay), and precompute ||m_c||^2 for all 1024 cells into d_ws.
// ---------------------------------------------------------------------------
__global__ void mmu_init_kernel(const int* __restrict__ node_cell,
                                const float* __restrict__ node_time,
                                const float* __restrict__ mmap,
                                float* __restrict__ out_ncell,
                                float* __restrict__ out_ntime,
                                float* __restrict__ ws_mnorm) {
  int i = blockIdx.x * blockDim.x + threadIdx.x;
  if (i < NB_NODES) {
    out_ncell[i] = (float)node_cell[i];
    out_ntime[i] = node_time[i];
  }
  if (i < NB_C) {
    const float* mp = mmap + (size_t)i * NB_D;
    float s = 0.f;
#pragma unroll 8
    for (int k = 0; k < NB_D; ++k) { float v = mp[k]; s += v * v; }
    ws_mnorm[i] = s;
  }
}

// ---------------------------------------------------------------------------
// Main fused kernel: cdist (WMMA f32 16x16x4) + row-argmin + scatter.
// One block = 32 rows x 1024 cols. 8 waves, each owns a 16x16 output tile,
// arranged 2 (row) x 4 (col); 16 column iterations, M tile double-buffered.
// ---------------------------------------------------------------------------
__global__ __launch_bounds__(256) void mmu_dist_kernel(
    const float* __restrict__ emb,      // [N, 128]
    const float* __restrict__ mmap,     // [1024, 128]
    const int*   __restrict__ nodes,    // [N]
    const float* __restrict__ times,    // [N]
    const float* __restrict__ ws_mnorm, // [1024]
    float* __restrict__ out_dist,       // [N, 1024]
    float* __restrict__ out_cells,      // [N]
    float* __restrict__ out_ncell,      // [NUM_NODES]
    float* __restrict__ out_ntime) {    // [NUM_NODES]

  __shared__ __align__(16) float Atile[ROWS_PB * LDS_STRIDE];       // 16.9 KB
  __shared__ __align__(16) float Mtile[2][COLS_PI * LDS_STRIDE];    // 67.6 KB
  __shared__ float xn[ROWS_PB];
  __shared__ float mn[NB_C];
  __shared__ unsigned long long rmin[ROWS_PB];

  const int tid  = threadIdx.x;
  const int row0 = blockIdx.x * ROWS_PB;

  // ---- stage A tile (32x128) into LDS, non-temporal (streamed once) ----
#pragma unroll
  for (int i = 0; i < 4; ++i) {
    int idx = tid + i * 256;            // 0..1023 float4 slots
    int r   = idx >> 5;                 // 32 float4 per row
    int c4  = idx & 31;
    v4f v = __builtin_nontemporal_load(
        (const v4f*)(emb + (size_t)(row0 + r) * NB_D + c4 * 4));
    *(v4f*)(&Atile[r * LDS_STRIDE + c4 * 4]) = v;
  }
  // ---- pull precomputed cell norms (4 KB, coalesced) ----
  for (int r = tid; r < NB_C; r += 256) mn[r] = ws_mnorm[r];
  __syncthreads();

  // ---- row norms from LDS + argmin slots init ----
  if (tid < ROWS_PB) {
    float s = 0.f;
#pragma unroll 8
    for (int k = 0; k < NB_D; ++k) { float v = Atile[tid * LDS_STRIDE + k]; s += v * v; }
    xn[tid]   = s;
    rmin[tid] = ~0ull;
  }
  __syncthreads();

  const int lane = tid & 31;
  const int wave = tid >> 5;
  const int wr   = (wave >> 2) * 16;    // wave row offset within block tile: 0 / 16
  const int wc   = (wave & 3) * 16;     // wave col offset within 64-col slab
  const int half = lane >> 4;           // K-pair select (ISA A/B layout)
  const int l15  = lane & 15;

  unsigned long long pmin[8];
#pragma unroll
  for (int r = 0; r < 8; ++r) pmin[r] = ~0ull;

  // prime the pipeline: tile 0 -> buffer 0
  stage_m_tile(mmap, &Mtile[0][0], 0, tid);

  for (int ci = 0; ci < ITERS; ++ci) {
    const int c0  = ci * COLS_PI;
    const int buf = ci & 1;

    wait_stage_done();   // tile ci now visible in Mtile[buf]

    // kick off async copy of tile ci+1 into the other buffer
    if (ci + 1 < ITERS)
      stage_m_tile(mmap, &Mtile[buf ^ 1][0], c0 + COLS_PI, tid);

    // ---- 16x16 tile GEMM: 32 x v_wmma_f32_16x16x4_f32 over K=128 ----
    v8f acc = {0.f, 0.f, 0.f, 0.f, 0.f, 0.f, 0.f, 0.f};
#pragma unroll
    for (int k = 0; k < NB_D / 4; ++k) {
      v2f a = *(const v2f*)(&Atile[(wr + l15) * LDS_STRIDE + k * 4 + half * 2]);
      v2f b = *(const v2f*)(&Mtile[buf][(wc + l15) * LDS_STRIDE + k * 4 + half * 2]);
      acc = __builtin_amdgcn_wmma_f32_16x16x4_f32(
          /*neg_a=*/false, a, /*neg_b=*/false, b,
          /*c_mod=*/(short)0, acc, /*reuse_a=*/false, /*reuse_b=*/false);
    }

    // ---- epilogue: distance, NT store, running packed argmin ----
    const int gcol = c0 + wc + l15;
    const float mnv = mn[gcol];
#pragma unroll
    for (int r = 0; r < 8; ++r) {
      int   lrow = wr + r + half * 8;   // 0..31 within block tile
      int   grow = row0 + lrow;
      float d2   = xn[lrow] + mnv - 2.f * acc[r];
      float dist = __builtin_sqrtf(fmaxf(d2, 0.f));
      __builtin_nontemporal_store(dist, out_dist + (size_t)grow * NB_C + gcol);
      unsigned long long p =
          ((unsigned long long)__float_as_uint(dist) << 32) | (unsigned)gcol;
      if (p < pmin[r]) pmin[r] = p;
    }
  }

  // ---- cross-lane argmin (16-lane halves), then ds_min_u64 into LDS ----
#pragma unroll
  for (int r = 0; r < 8; ++r) {
    unsigned long long p = pmin[r];
#pragma unroll
    for (int m = 1; m <= 8; m <<= 1) {
      unsigned long long q = __shfl_xor(p, m, 32);
      if (q < p) p = q;
    }
    if (l15 == 0) atomicMin(&rmin[wr + r + half * 8], p);
  }
  __syncthreads();

  // ---- emit cells + scatter node tables ----
  if (tid < ROWS_PB) {
    unsigned long long p = rmin[tid];
    int cell = (int)(unsigned)(p & 0xFFFFFFFFull);
    int grow = row0 + tid;
    out_cells[grow] = (float)cell;
    int node = nodes[grow];
    out_ncell[node] = (float)cell;
    out_ntime[node] = times[grow];
  }
}

// ---------------------------------------------------------------------------
extern "C" void kernel_launch(void* const* d_in, const int* in_sizes, int n_in,
                              void* d_out, int out_size, void* d_ws, size_t ws_size,
                              hipStream_t stream) {
  const float* emb       = (const float*)d_in[0];
  const float* mmap      = (const float*)d_in[1];
  const int*   nodes     = (const int*)d_in[2];
  const float* times     = (const float*)d_in[3];
  const int*   node_cell = (const int*)d_in[4];
  const float* node_time = (const float*)d_in[5];

  float* out       = (float*)d_out;
  float* out_dist  = out;                              // N*C
  float* out_cells = out_dist + (size_t)NB_N * NB_C;   // N
  float* out_ncell = out_cells + NB_N;                 // NUM_NODES
  float* out_ntime = out_ncell + NB_NODES;             // NUM_NODES
  float* ws_mnorm  = (float*)d_ws;                     // 1024 floats

  mmu_init_kernel<<<(NB_NODES + 255) / 256, 256, 0, stream>>>(
      node_cell, node_time, mmap, out_ncell, out_ntime, ws_mnorm);

  mmu_dist_kernel<<<NB_N / ROWS_PB, 256, 0, stream>>>(
      emb, mmap, nodes, times, ws_mnorm,
      out_dist, out_cells, out_ncell, out_ntime);
}